// LUTCompatibility_48318382080004
// MI455X (gfx1250) — compile-verified
//
#include <hip/hip_runtime.h>
#include <math.h>

// ---------------- problem constants (from reference init_kwargs) ----------------
#define NBX 168
#define NBY 480
#define NBL 6
#define EXT 2                         // stddev_trunc 2.5 -> 5x5 window
#define WIN 5
#define INV_SQRT2 0.70710678118654752440f

// ---------------- tiling ----------------
#define TILE 16
#define HALO EXT
#define TDIM (TILE + 2*HALO)          // 20
#define TCELLS (TDIM*TDIM*NBL)        // 2400 floats = 9.6 KB LDS
#define NTX ((NBX + TILE - 1)/TILE)   // 11
#define NTY ((NBY + TILE - 1)/TILE)   // 30
#define NTILES (NTX*NTY)              // 330
// Padded demand map: every tile's 20x20 halo region is fully in-bounds.
#define NBXP (NTX*TILE + 2*HALO)      // 180
#define NBYP (NTY*TILE + 2*HALO)      // 484
#define DEMPSZ (NBXP*NBYP*NBL)        // 522720 floats ~= 2.09 MB (L2-resident)
#define ROWLEN (TDIM*NBL)             // 120 contiguous floats per tile row
#define ROWSTRIDE (NBYP*NBL)          // 2904 elements between rows

typedef unsigned int u32x4 __attribute__((ext_vector_type(4)));
typedef int          i32x4 __attribute__((ext_vector_type(4)));
typedef int          i32x8 __attribute__((ext_vector_type(8)));

__device__ __forceinline__ void wait_tensorcnt0() {
#if __has_builtin(__builtin_amdgcn_s_wait_tensorcnt)
    __builtin_amdgcn_s_wait_tensorcnt(0);
#else
    asm volatile("s_wait_tensorcnt 0x0" ::: "memory");
#endif
}

#if __has_builtin(__builtin_amdgcn_tensor_load_to_lds)
#define HAVE_TDM 1
// One 2D TDM descriptor: `rows` rows of `row_elems` contiguous f32, global row
// stride `row_stride_elems`, into LDS at byte offset lds_addr. (ISA 08 §8)
__device__ __forceinline__ void tdm_load_2d(unsigned lds_addr, const float* gptr,
                                            unsigned row_elems, unsigned rows,
                                            unsigned row_stride_elems) {
    unsigned long long ga = (unsigned long long)(uintptr_t)gptr;
    u32x4 g0 = { 1u,                                   // count=1 (valid user D#)
                 lds_addr,                             // lds_addr [63:32]
                 (unsigned)(ga & 0xFFFFFFFFu),         // global_addr lo
                 (unsigned)((ga >> 32) & 0x01FFFFFFu) | 0x80000000u }; // addr hi | type=2
    i32x8 g1 = { (int)(2u << 16),                          // data_size=4B; mask=0
                 (int)((row_elems & 0xFFFFu) << 16),       // tensor_dim0 lo16
                 (int)(((row_elems >> 16) & 0xFFFFu) | ((rows & 0xFFFFu) << 16)),
                 (int)(((rows >> 16) & 0xFFFFu) | ((row_elems & 0xFFFFu) << 16)), // tile_dim0
                 (int)(rows & 0xFFFFu),                    // tile_dim1 (tile_dim2=0 -> 2D)
                 (int)row_stride_elems,                    // tensor_dim0_stride lo32
                 0, 0 };
    i32x4 gz = { 0, 0, 0, 0 };
#if defined(__clang_major__) && (__clang_major__ >= 23)
    i32x8 gz8 = { 0, 0, 0, 0, 0, 0, 0, 0 };
    __builtin_amdgcn_tensor_load_to_lds(g0, g1, gz, gz, gz8, 0);
#else
    __builtin_amdgcn_tensor_load_to_lds(g0, g1, gz, gz, 0);
#endif
}
#else
#define HAVE_TDM 0
#endif

// Per-axis truncated-Gaussian bin weights: w_j = 0.5*(erf(edge_{j+1}) - erf(edge_j)),
// edges at (b0 + j - EXT - c)/sqrt(2), zeroed for out-of-range bins (matches reference).
__device__ __forceinline__ void axis_weights(float c, int nbins, int& b0, float w[WIN]) {
    float f = floorf(c);
    b0 = (int)f;
    float e[WIN + 1];
#pragma unroll
    for (int j = 0; j < WIN + 1; ++j)
        e[j] = erff((f + (float)(j - EXT) - c) * INV_SQRT2);
#pragma unroll
    for (int j = 0; j < WIN; ++j) {
        int b = b0 + j - EXT;
        w[j] = (b >= 0 && b < nbins) ? 0.5f * (e[j + 1] - e[j]) : 0.0f;
    }
}

// K1: zero padded demand map, tile counters, and output.
__global__ void k_zero(float* __restrict__ demP, int* __restrict__ tileCount,
                       int* __restrict__ tileCursor, float* __restrict__ out, int nOut) {
    int i = blockIdx.x * blockDim.x + threadIdx.x;
    int stride = gridDim.x * blockDim.x;
    for (int j = i; j < DEMPSZ; j += stride) demP[j] = 0.0f;
    for (int j = i; j < NTILES; j += stride) { tileCount[j] = 0; tileCursor[j] = 0; }
    for (int j = i; j < nOut; j += stride) __builtin_nontemporal_store(0.0f, &out[j]);
}

// K2: gather node data once, compute instance centers, histogram into tiles.
__global__ void k_centers(const float* __restrict__ pos, const float* __restrict__ nsx,
                          const float* __restrict__ nsy, const int* __restrict__ lutIdx,
                          const int* __restrict__ lutType, int M, int N,
                          float* __restrict__ tmpCx, float* __restrict__ tmpCy,
                          int* __restrict__ tmpPk, int* __restrict__ tileCount) {
    int i = blockIdx.x * blockDim.x + threadIdx.x;
    if (i >= M) return;
    int idx = lutIdx[i];
    float cx = pos[idx] + 0.5f * nsx[idx];
    float cy = pos[N + idx] + 0.5f * nsy[idx];
    int t = lutType[idx];
    tmpCx[i] = cx;
    tmpCy[i] = cy;
    tmpPk[i] = (idx << 3) | t;                 // node index (<2^21) + type (3 bits)
    int tx = ((int)floorf(cx)) >> 4;           // b0x in [0,166] -> tile 0..10
    int ty = ((int)floorf(cy)) >> 4;           // b0y in [0,478] -> tile 0..29
    atomicAdd(&tileCount[tx * NTY + ty], 1);
}

// K3: exclusive prefix sum over 330 tile counts (trivial; single thread).
__global__ void k_scan(const int* __restrict__ tileCount, int* __restrict__ tileStart) {
    if (blockIdx.x == 0 && threadIdx.x == 0) {
        int acc = 0;
        for (int t = 0; t < NTILES; ++t) { tileStart[t] = acc; acc += tileCount[t]; }
        tileStart[NTILES] = acc;
    }
}

// K4: bucket-reorder instance payloads by tile (1 global atomic per instance).
__global__ void k_reorder(const float* __restrict__ tmpCx, const float* __restrict__ tmpCy,
                          const int* __restrict__ tmpPk, const int* __restrict__ tileStart,
                          int* __restrict__ tileCursor, float* __restrict__ ordCx,
                          float* __restrict__ ordCy, int* __restrict__ ordPk, int M) {
    int i = blockIdx.x * blockDim.x + threadIdx.x;
    if (i >= M) return;
    float cx = tmpCx[i], cy = tmpCy[i];
    int t = (((int)floorf(cx)) >> 4) * NTY + (((int)floorf(cy)) >> 4);
    int p = tileStart[t] + atomicAdd(&tileCursor[t], 1);
    ordCx[p] = cx;
    ordCy[p] = cy;
    ordPk[p] = tmpPk[i];
}

// K5: one workgroup per tile; accumulate all splats in LDS (ds_add_f32),
// then flush <=2400 cells with global atomics (vs 25 per instance naive).
__global__ __launch_bounds__(256) void k_accum(const float* __restrict__ ordCx,
                                               const float* __restrict__ ordCy,
                                               const int* __restrict__ ordPk,
                                               const int* __restrict__ tileStart,
                                               float* __restrict__ demP) {
    __shared__ float acc[TCELLS];
    int tile = blockIdx.x;
    int tx = tile / NTY, ty = tile % NTY;
    int gx0 = tx * TILE - HALO;
    int gy0 = ty * TILE - HALO;
    for (int j = threadIdx.x; j < TCELLS; j += blockDim.x) acc[j] = 0.0f;
    __syncthreads();
    int s = tileStart[tile], e = tileStart[tile + 1];
    for (int p = s + (int)threadIdx.x; p < e; p += blockDim.x) {
        float cx = ordCx[p], cy = ordCy[p];
        int t = ordPk[p] & 7;
        int bx0, by0; float wx[WIN], wy[WIN];
        axis_weights(cx, NBX, bx0, wx);
        axis_weights(cy, NBY, by0, wy);
#pragma unroll
        for (int jx = 0; jx < WIN; ++jx) {
            if (wx[jx] == 0.0f) continue;                 // out-of-range bin: exact zero
            int lx = bx0 + jx - EXT - gx0;                // in [0, TDIM)
#pragma unroll
            for (int jy = 0; jy < WIN; ++jy) {
                float w = wx[jx] * wy[jy];
                if (w == 0.0f) continue;
                int ly = by0 + jy - EXT - gy0;
                atomicAdd(&acc[(lx * TDIM + ly) * NBL + t], w);  // LDS ds_add_f32
            }
        }
    }
    __syncthreads();
    // Padded map: every tile cell is in-bounds -> branch-free flush.
    for (int j = threadIdx.x; j < TCELLS; j += blockDim.x) {
        float v = acc[j];
        if (v == 0.0f) continue;
        int t  = j % NBL;
        int ly = (j / NBL) % TDIM;
        int lx = j / (NBL * TDIM);
        int px = tx * TILE + lx;     // padded coords
        int py = ty * TILE + ly;
        atomicAdd(&demP[(px * NBYP + py) * NBL + t], v);
    }
}

// K6: one workgroup per tile; TDM-stage the dem tile(+halo) into LDS with one
// tensor_load_to_lds descriptor, then each instance dots its 25 weights
// against LDS and writes its area (NT store).
__global__ __launch_bounds__(256) void k_area(const float* __restrict__ ordCx,
                                              const float* __restrict__ ordCy,
                                              const int* __restrict__ ordPk,
                                              const int* __restrict__ tileStart,
                                              const float* __restrict__ demP,
                                              float* __restrict__ out) {
    __shared__ float tileD[TCELLS];
    int tile = blockIdx.x;
    int tx = tile / NTY, ty = tile % NTY;
    int gx0 = tx * TILE - HALO;
    int gy0 = ty * TILE - HALO;
    const float* gbase = demP + ((size_t)(tx * TILE) * NBYP + (size_t)(ty * TILE)) * NBL;
#if HAVE_TDM
    if (threadIdx.x < 32) {          // wave 0 issues the tensor DMA (TENSORcnt)
        unsigned ldsOff = (unsigned)(uintptr_t)(void*)tileD;  // low 32 bits = LDS byte addr
        tdm_load_2d(ldsOff, gbase, ROWLEN, TDIM, ROWSTRIDE);
        wait_tensorcnt0();
    }
#else
    for (int j = threadIdx.x; j < TCELLS; j += blockDim.x) {
        int r = j / ROWLEN, cIdx = j % ROWLEN;
        tileD[j] = gbase[(size_t)r * ROWSTRIDE + cIdx];
    }
#endif
    __syncthreads();
    int s = tileStart[tile], e = tileStart[tile + 1];
    for (int p = s + (int)threadIdx.x; p < e; p += blockDim.x) {
        float cx = ordCx[p], cy = ordCy[p];
        int pk = ordPk[p];
        int t = pk & 7, nodeIdx = pk >> 3;
        int bx0, by0; float wx[WIN], wy[WIN];
        axis_weights(cx, NBX, bx0, wx);
        axis_weights(cy, NBY, by0, wy);
        float sum = 0.0f;
#pragma unroll
        for (int jx = 0; jx < WIN; ++jx) {
            int lx = bx0 + jx - EXT - gx0;
            float partial = 0.0f;
#pragma unroll
            for (int jy = 0; jy < WIN; ++jy) {
                int ly = by0 + jy - EXT - gy0;
                partial += wy[jy] * tileD[(lx * TDIM + ly) * NBL + t];
            }
            sum += wx[jx] * partial;        // zero weights kill any halo contribution
        }
        __builtin_nontemporal_store(sum * (1.0f / 16.0f), &out[nodeIdx]);
    }
}

extern "C" void kernel_launch(void* const* d_in, const int* in_sizes, int n_in,
                              void* d_out, int out_size, void* d_ws, size_t ws_size,
                              hipStream_t stream) {
    const float* pos     = (const float*)d_in[0];
    const float* nsx     = (const float*)d_in[1];
    const float* nsy     = (const float*)d_in[2];
    const int*   lutIdx  = (const int*)d_in[3];
    const int*   lutType = (const int*)d_in[4];
    float* out = (float*)d_out;
    const int N = in_sizes[1];   // nodes (1.5M)
    const int M = in_sizes[3];   // LUT instances (1M)

    // workspace layout (all 4-byte elements)
    float* demP      = (float*)d_ws;                 // DEMPSZ (padded map)
    int*   tileCount = (int*)(demP + DEMPSZ);        // NTILES
    int*   tileStart = tileCount + NTILES;           // NTILES + 1
    int*   tileCursor= tileStart + NTILES + 1;       // NTILES
    float* tmpCx     = (float*)(tileCursor + NTILES);
    float* tmpCy     = tmpCx + M;
    int*   tmpPk     = (int*)(tmpCy + M);
    float* ordCx     = (float*)(tmpPk + M);
    float* ordCy     = ordCx + M;
    int*   ordPk     = (int*)(ordCy + M);

    const int B = 256;
    const int gM = (M + B - 1) / B;

    k_zero<<<1024, B, 0, stream>>>(demP, tileCount, tileCursor, out, out_size);
    k_centers<<<gM, B, 0, stream>>>(pos, nsx, nsy, lutIdx, lutType, M, N,
                                    tmpCx, tmpCy, tmpPk, tileCount);
    k_scan<<<1, 32, 0, stream>>>(tileCount, tileStart);
    k_reorder<<<gM, B, 0, stream>>>(tmpCx, tmpCy, tmpPk, tileStart, tileCursor,
                                    ordCx, ordCy, ordPk, M);
    k_accum<<<NTILES, B, 0, stream>>>(ordCx, ordCy, ordPk, tileStart, demP);
    k_area<<<NTILES, B, 0, stream>>>(ordCx, ordCy, ordPk, tileStart, demP, out);
}